// SelectiveSSM_21260088115344
// MI455X (gfx1250) — compile-verified
//
#include <hip/hip_runtime.h>
#include <hip/hip_bf16.h>
#include <math.h>
#include <stdint.h>

// ---- problem constants (match reference) ----
#define BATCH   2
#define SEQLEN  2048
#define DMODEL  1024
#define DSTATE  16
#define DCONV   4
#define DTRANK  64
#define PDIM    (DTRANK + 2 * DSTATE)   // 96
#define MTOT    (BATCH * SEQLEN)        // 4096

#define CHUNK   64                      // timesteps staged per LDS buffer
#define NCHUNK  (SEQLEN / CHUNK)        // 32

typedef __attribute__((ext_vector_type(2))) float v2f;
typedef __attribute__((ext_vector_type(8))) float v8f;

// ---------------------------------------------------------------------------
// Kernel 1: causal depthwise conv1d.  x[b,l,d] = sum_j u[b,l-3+j,d]*w[d,j] + cb[d]
// ---------------------------------------------------------------------------
__global__ void ssm_conv1d(const float* __restrict__ u,
                           const float* __restrict__ w,   // (D,1,4)
                           const float* __restrict__ cb,  // (D,)
                           float* __restrict__ x) {
    int idx = blockIdx.x * blockDim.x + threadIdx.x;      // over MTOT*DMODEL
    int d = idx & (DMODEL - 1);
    int m = idx >> 10;                                    // row in [0, MTOT)
    int l = m & (SEQLEN - 1);
    float acc = cb[d];
#pragma unroll
    for (int j = 0; j < DCONV; ++j) {
        int li = l - (DCONV - 1) + j;
        float uv = (li >= 0) ? u[(m - l + li) * DMODEL + d] : 0.f;
        acc += uv * w[d * DCONV + j];
    }
    x[idx] = acc;
}

// ---------------------------------------------------------------------------
// Kernel 2: proj = x @ x_proj_w^T   (M=4096, K=1024, P=96) via fp32 WMMA 16x16x4.
// ---------------------------------------------------------------------------
__global__ void ssm_xproj_wmma(const float* __restrict__ x,
                               const float* __restrict__ w,   // (96,1024) row-major
                               float* __restrict__ proj) {    // (M, 96)
    int wid  = (blockIdx.x * blockDim.x + threadIdx.x) >> 5;
    int lane = threadIdx.x & 31;
    int tm = wid / (PDIM / 16);
    int tp = wid % (PDIM / 16);
    int lr = lane & 15;
    int hi = lane >> 4;
    int kh = hi << 1;                                     // 0 or 2
    const float* arow = x + (size_t)(tm * 16 + lr) * DMODEL + kh;
    const float* brow = w + (size_t)(tp * 16 + lr) * DMODEL + kh;
    v8f acc = {};
    for (int k0 = 0; k0 < DMODEL; k0 += 4) {
        v2f a  = *(const v2f*)(arow + k0);
        v2f bm = *(const v2f*)(brow + k0);
        acc = __builtin_amdgcn_wmma_f32_16x16x4_f32(
            false, a, false, bm, (short)0, acc, false, false);
    }
#pragma unroll
    for (int r = 0; r < 8; ++r) {
        int row = tm * 16 + r + hi * 8;                   // C/D: VGPR r -> M=r / M=r+8
        proj[(size_t)row * PDIM + tp * 16 + lr] = acc[r];
    }
}

// ---------------------------------------------------------------------------
// Kernel 3: delta = softplus(dt @ dt_proj_w^T + bias)  (M=4096, K=64, N=1024).
// ---------------------------------------------------------------------------
__global__ void ssm_dtproj_wmma(const float* __restrict__ proj,   // (M,96)
                                const float* __restrict__ w,      // (1024,64) row-major
                                const float* __restrict__ bias,   // (1024,)
                                float* __restrict__ delta) {      // (M,1024)
    int wid  = (blockIdx.x * blockDim.x + threadIdx.x) >> 5;
    int lane = threadIdx.x & 31;
    int tm = wid / (DMODEL / 16);
    int td = wid % (DMODEL / 16);
    int lr = lane & 15;
    int hi = lane >> 4;
    int kh = hi << 1;
    const float* arow = proj + (size_t)(tm * 16 + lr) * PDIM + kh;   // dt cols 0..63
    const float* brow = w    + (size_t)(td * 16 + lr) * DTRANK + kh;
    v8f acc = {};
#pragma unroll
    for (int k0 = 0; k0 < DTRANK; k0 += 4) {
        v2f a  = *(const v2f*)(arow + k0);
        v2f bm = *(const v2f*)(brow + k0);
        acc = __builtin_amdgcn_wmma_f32_16x16x4_f32(
            false, a, false, bm, (short)0, acc, false, false);
    }
    int col = td * 16 + lr;
    float bv = bias[col];
#pragma unroll
    for (int r = 0; r < 8; ++r) {
        int row = tm * 16 + r + hi * 8;
        float v = acc[r] + bv;
        float sp = (v > 20.f) ? v : log1pf(expf(v));      // softplus, overflow-safe
        delta[(size_t)row * DMODEL + col] = sp;
    }
}

// ---------------------------------------------------------------------------
// async copy helpers (CDNA5 ASYNCcnt path)
// ---------------------------------------------------------------------------
__device__ __forceinline__ void async_b128(uint32_t lds_byte_off, const void* gptr) {
    unsigned long long g = (unsigned long long)(uintptr_t)gptr;
    asm volatile("global_load_async_to_lds_b128 %0, %1, off"
                 :: "v"(lds_byte_off), "v"(g) : "memory");
}
__device__ __forceinline__ void wait_async_le4() {
    asm volatile("s_wait_asynccnt 0x4" ::: "memory");
}
__device__ __forceinline__ void wait_async_0() {
    asm volatile("s_wait_asynccnt 0x0" ::: "memory");
}
__device__ __forceinline__ uint32_t lds_off(const void* p) {
    // generic LDS pointers are {SHARED_BASE_hi32, lds_offset[31:0]} -> truncate
    return (uint32_t)(uintptr_t)p;
}

// ---------------------------------------------------------------------------
// Kernel 4: selective scan with double-buffered async LDS staging.
// Block = 256 threads = 16 channels (16 lanes/channel, one per state n).
// Per chunk of 64 timesteps each thread issues exactly one
// global_load_async_to_lds_b128 per array (delta,u,B,C): 16KB/chunk, 2 buffers.
// State h stays in a register for all 2048 steps.
// ---------------------------------------------------------------------------
__global__ void ssm_scan(const float* __restrict__ delta,  // (B,L,D)
                         const float* __restrict__ proj,   // (B,L,96): [.,64+n]=B, [.,80+n]=C
                         const float* __restrict__ u,      // (B,L,D)
                         const float* __restrict__ A_log,  // (D,N)
                         const float* __restrict__ Dskip,  // (D,)
                         float* __restrict__ out) {        // (B,L,D)
    __shared__ float sD[2][CHUNK * 16];
    __shared__ float sU[2][CHUNK * 16];
    __shared__ float sB[2][CHUNK * 16];
    __shared__ float sC[2][CHUNK * 16];

    const int tid  = threadIdx.x;
    const int lane = tid & 31;
    const int n    = lane & 15;               // state index
    const int cloc = (tid >> 5) * 2 + (lane >> 4);   // channel within block [0,16)
    const int ch0  = blockIdx.x * 16;         // base channel; whole block same batch
    const int b    = ch0 >> 10;
    const int d0   = ch0 & (DMODEL - 1);
    const int d    = d0 + cloc;

    // staging role: thread tid covers row t = tid/4, quarter q = tid%4 (4 floats)
    const int st = tid >> 2;
    const int sq = (tid & 3) * 4;

    const float A   = -expf(A_log[d * DSTATE + n]);   // A < 0 always
    const float dsk = Dskip[d];
    float h = 0.f;

    const size_t rowD = (size_t)(b * SEQLEN) * DMODEL;
    const size_t rowP = (size_t)(b * SEQLEN) * PDIM;

    // issue chunk 0
    {
        size_t gr = rowD + (size_t)st * DMODEL + d0 + sq;
        size_t pr = rowP + (size_t)st * PDIM;
        uint32_t lo = (uint32_t)((st * 16 + sq) * 4);
        async_b128(lds_off(sD[0]) + lo, delta + gr);
        async_b128(lds_off(sU[0]) + lo, u + gr);
        async_b128(lds_off(sB[0]) + lo, proj + pr + DTRANK + sq);
        async_b128(lds_off(sC[0]) + lo, proj + pr + DTRANK + DSTATE + sq);
    }

    for (int c = 0; c < NCHUNK; ++c) {
        const int buf = c & 1;
        if (c + 1 < NCHUNK) {           // prefetch next chunk into other buffer
            int t0n = (c + 1) * CHUNK;
            size_t gr = rowD + (size_t)(t0n + st) * DMODEL + d0 + sq;
            size_t pr = rowP + (size_t)(t0n + st) * PDIM;
            uint32_t lo = (uint32_t)((st * 16 + sq) * 4);
            int nb = buf ^ 1;
            async_b128(lds_off(sD[nb]) + lo, delta + gr);
            async_b128(lds_off(sU[nb]) + lo, u + gr);
            async_b128(lds_off(sB[nb]) + lo, proj + pr + DTRANK + sq);
            async_b128(lds_off(sC[nb]) + lo, proj + pr + DTRANK + DSTATE + sq);
            wait_async_le4();           // in-order: current chunk's 4 have landed
        } else {
            wait_async_0();
        }
        __syncthreads();                // staged data visible to all waves

        const int t0 = c * CHUNK;
        float* yo = out + rowD + (size_t)t0 * DMODEL + d;
#pragma unroll 4
        for (int t = 0; t < CHUNK; ++t) {
            float dt = sD[buf][t * 16 + cloc];
            float ut = sU[buf][t * 16 + cloc];
            float bt = sB[buf][t * 16 + n];
            float ct = sC[buf][t * 16 + n];
            float dA   = expf(dt * A);
            float Bbar = (dA - 1.f) * bt / A;
            h = dA * h + Bbar * ut;
            float yp = h * ct;
            yp += __shfl_xor(yp, 8, 16);
            yp += __shfl_xor(yp, 4, 16);
            yp += __shfl_xor(yp, 2, 16);
            yp += __shfl_xor(yp, 1, 16);
            if (n == 0) yo[(size_t)t * DMODEL] = yp + dsk * ut;
        }
        __syncthreads();                // all reads done before buffer is reused
    }
}

// ---------------------------------------------------------------------------
extern "C" void kernel_launch(void* const* d_in, const int* in_sizes, int n_in,
                              void* d_out, int out_size, void* d_ws, size_t ws_size,
                              hipStream_t stream) {
    (void)in_sizes; (void)n_in; (void)out_size; (void)ws_size;
    const float* u       = (const float*)d_in[0];
    const float* conv_w  = (const float*)d_in[1];
    const float* conv_b  = (const float*)d_in[2];
    const float* xproj_w = (const float*)d_in[3];
    const float* dt_w    = (const float*)d_in[4];
    const float* dt_b    = (const float*)d_in[5];
    const float* A_log   = (const float*)d_in[6];
    const float* D_skip  = (const float*)d_in[7];
    float* out = (float*)d_out;

    // workspace layout (bytes): x @0 (16MB), proj @16MB (1.5MB), delta @18MB (16MB)
    float* x     = (float*)d_ws;
    float* proj  = (float*)((char*)d_ws + ((size_t)16 << 20));
    float* delta = (float*)((char*)d_ws + ((size_t)18 << 20));

    ssm_conv1d<<<(MTOT * DMODEL) / 256, 256, 0, stream>>>(u, conv_w, conv_b, x);
    ssm_xproj_wmma<<<(MTOT / 16) * (PDIM / 16) / 4, 128, 0, stream>>>(x, xproj_w, proj);
    ssm_dtproj_wmma<<<(MTOT / 16) * (DMODEL / 16) / 4, 128, 0, stream>>>(proj, dt_w, dt_b, delta);
    ssm_scan<<<(BATCH * DMODEL) / 16, 256, 0, stream>>>(delta, proj, u, A_log, D_skip, out);
}